// GATModel_36541581754798
// MI455X (gfx1250) — compile-verified
//
#include <hip/hip_runtime.h>
#include <hip/hip_bf16.h>
#include <math.h>

#define N_NODES  30000
#define N_EDGES  480000
#define N_GRAPHS 64
#define IN_DIM   128
#define HID      64
#define E_TOT    (N_EDGES + N_NODES)   // edges + self loops
#define NEG_SLOPE 0.2f

typedef float v2f __attribute__((ext_vector_type(2)));
typedef float v8f __attribute__((ext_vector_type(8)));

// ---------------------------------------------------------------------------
// float atomic max via ordered-int trick (works for any sign, init = -inf)
// ---------------------------------------------------------------------------
__device__ __forceinline__ void atomic_max_f32(float* addr, float val) {
  if (val >= 0.0f) {
    atomicMax((int*)addr, __float_as_int(val));
  } else {
    atomicMin((unsigned int*)addr, (unsigned int)__float_as_int(val));
  }
}

// ---------------------------------------------------------------------------
// C[M,N] = A[M,K] * B[K,N]   fp32, one 16x16 tile per wave32 via
// V_WMMA_F32_16X16X4_F32 (K stepped by 4).
// A-tile lane layout (ISA 7.12.2): lanes 0-15 -> M=0..15 / K={0,1};
// lanes 16-31 -> same M, K={2,3}. B mirrored. C/D: VGPR v -> row v+8*(lane>>4).
// ---------------------------------------------------------------------------
template <int K, int N>
__global__ __launch_bounds__(256) void gemm_wmma_f32(
    const float* __restrict__ A, const float* __restrict__ B,
    float* __restrict__ C, int M) {
  const int wave = (int)((blockIdx.x * blockDim.x + threadIdx.x) >> 5);
  const int lane = threadIdx.x & 31;
  constexpr int NT = N / 16;
  const int tm = (wave / NT) * 16;
  const int tn = (wave % NT) * 16;
  if (tm >= M) return;                       // wave-uniform exit: EXEC stays all-1s
  const int l15  = lane & 15;
  const int koff = (lane >> 4) << 1;         // 0 or 2
  const float* arow = A + (size_t)(tm + l15) * K + koff;
  const float* bcol = B + (size_t)koff * N + tn + l15;
  v8f acc = {};
#pragma unroll 8
  for (int kk = 0; kk < K; kk += 4) {
    v2f a = *(const v2f*)(arow + kk);        // contiguous 8B load
    v2f b;
    b.x = bcol[(size_t)kk * N];
    b.y = bcol[(size_t)kk * N + N];
    acc = __builtin_amdgcn_wmma_f32_16x16x4_f32(
        /*neg_a=*/false, a, /*neg_b=*/false, b,
        /*c_mod=*/(short)0, acc, /*reuse_a=*/false, /*reuse_b=*/false);
  }
  const int mbase = tm + ((lane >> 4) << 3);
  float* crow = C + (size_t)mbase * N + tn + l15;
#pragma unroll
  for (int v = 0; v < 8; ++v) crow[(size_t)v * N] = acc[v];
}

// ---------------------------------------------------------------------------
// per-(node,head) attention scores: a_src[i] = <h[i], att_src[h]>, same for dst
// h layout: [node][H][HID] flattened, so row i = node*H + head starts at i*HID.
// ---------------------------------------------------------------------------
template <int H>
__global__ void att_scores(const float* __restrict__ h,
                           const float* __restrict__ ws,
                           const float* __restrict__ wd,
                           float* __restrict__ as, float* __restrict__ ad) {
  int i = blockIdx.x * blockDim.x + threadIdx.x;
  if (i >= N_NODES * H) return;
  int hh = i % H;
  const float* hp = h + (size_t)i * HID;
  const float* ps = ws + hh * HID;
  const float* pd = wd + hh * HID;
  float s = 0.f, d = 0.f;
#pragma unroll
  for (int c = 0; c < HID; ++c) { float v = hp[c]; s += v * ps[c]; d += v * pd[c]; }
  as[i] = s;
  ad[i] = d;
}

__global__ void fill_f32(float* __restrict__ p, float val, int n) {
  int i = blockIdx.x * blockDim.x + threadIdx.x;
  if (i < n) p[i] = val;
}

// ---------------------------------------------------------------------------
// edge pass 1: segment-max of leaky_relu(a_src[s] + a_dst[d]) over dst
// ---------------------------------------------------------------------------
template <int H>
__global__ void edge_lrelu_max(const int* __restrict__ src, const int* __restrict__ dst,
                               const float* __restrict__ as, const float* __restrict__ ad,
                               float* __restrict__ emax) {
  int e = blockIdx.x * blockDim.x + threadIdx.x;
  if (e >= E_TOT) return;
  int s = (e < N_EDGES) ? src[e] : (e - N_EDGES);
  int d = (e < N_EDGES) ? dst[e] : (e - N_EDGES);
#pragma unroll
  for (int h = 0; h < H; ++h) {
    float v = as[s * H + h] + ad[d * H + h];
    v = (v > 0.f) ? v : NEG_SLOPE * v;
    atomic_max_f32(&emax[d * H + h], v);
  }
}

// ---------------------------------------------------------------------------
// edge pass 2: ev = exp(v - emax[dst]); denom[dst] += ev
// ---------------------------------------------------------------------------
template <int H>
__global__ void edge_exp_sum(const int* __restrict__ src, const int* __restrict__ dst,
                             const float* __restrict__ as, const float* __restrict__ ad,
                             const float* __restrict__ emax,
                             float* __restrict__ ev, float* __restrict__ denom) {
  int e = blockIdx.x * blockDim.x + threadIdx.x;
  if (e >= E_TOT) return;
  int s = (e < N_EDGES) ? src[e] : (e - N_EDGES);
  int d = (e < N_EDGES) ? dst[e] : (e - N_EDGES);
#pragma unroll
  for (int h = 0; h < H; ++h) {
    float v = as[s * H + h] + ad[d * H + h];
    v = (v > 0.f) ? v : NEG_SLOPE * v;
    float x = expf(v - emax[d * H + h]);
    ev[e * H + h] = x;
    atomicAdd(&denom[d * H + h], x);
  }
}

// ---------------------------------------------------------------------------
// edge pass 3: out[dst] += (ev/denom[dst]) * h[src]   (one thread per e,h,c)
// ---------------------------------------------------------------------------
template <int H>
__global__ void edge_msg(const int* __restrict__ src, const int* __restrict__ dst,
                         const float* __restrict__ ev, const float* __restrict__ denom,
                         const float* __restrict__ h, float* __restrict__ out) {
  int idx = blockIdx.x * blockDim.x + threadIdx.x;
  if (idx >= E_TOT * H * HID) return;
  int c  = idx & (HID - 1);
  int eh = idx / HID;
  int hh = eh % H;
  int e  = eh / H;
  int s = (e < N_EDGES) ? src[e] : (e - N_EDGES);
  int d = (e < N_EDGES) ? dst[e] : (e - N_EDGES);
  float alpha = ev[e * H + hh] / denom[d * H + hh];
  atomicAdd(&out[((size_t)d * H + hh) * HID + c],
            alpha * h[((size_t)s * H + hh) * HID + c]);
}

template <int C>
__global__ void bias_relu(float* __restrict__ o, const float* __restrict__ b, int n) {
  int i = blockIdx.x * blockDim.x + threadIdx.x;
  if (i >= n * C) return;
  float v = o[i] + b[i % C];
  o[i] = (v > 0.f) ? v : 0.f;
}

__global__ void pool_max(const float* __restrict__ o, const int* __restrict__ batch,
                         float* __restrict__ g) {
  int i = blockIdx.x * blockDim.x + threadIdx.x;
  if (i >= N_NODES * HID) return;
  int node = i >> 6;       // / HID
  int c    = i & (HID - 1);
  atomic_max_f32(&g[batch[node] * HID + c], o[i]);
}

// ---------------------------------------------------------------------------
// FC head: relu(g @ w1 + b1) @ w2 + b2 -> log_softmax.  One block per graph.
// ---------------------------------------------------------------------------
__global__ void head_kernel(const float* __restrict__ g,
                            const float* __restrict__ w1, const float* __restrict__ b1,
                            const float* __restrict__ w2, const float* __restrict__ b2,
                            float* __restrict__ out) {
  __shared__ float sg[HID];
  __shared__ float sh[HID];
  __shared__ float logits[2];
  const int gi = blockIdx.x;
  const int t  = threadIdx.x;
  sg[t] = g[gi * HID + t];
  __syncthreads();
  float acc = b1[t];
#pragma unroll
  for (int k = 0; k < HID; ++k) acc += sg[k] * w1[k * HID + t];
  sh[t] = (acc > 0.f) ? acc : 0.f;
  __syncthreads();
  if (t < 2) {
    float a2 = b2[t];
#pragma unroll
    for (int k = 0; k < HID; ++k) a2 += sh[k] * w2[k * 2 + t];
    logits[t] = a2;
  }
  __syncthreads();
  if (t < 2) {
    float m   = fmaxf(logits[0], logits[1]);
    float lse = m + logf(expf(logits[0] - m) + expf(logits[1] - m));
    out[gi * 2 + t] = logits[t] - lse;
  }
}

// ---------------------------------------------------------------------------
extern "C" void kernel_launch(void* const* d_in, const int* in_sizes, int n_in,
                              void* d_out, int out_size, void* d_ws, size_t ws_size,
                              hipStream_t stream) {
  (void)in_sizes; (void)n_in; (void)out_size; (void)ws_size;
  const float* x    = (const float*)d_in[0];
  const int*   ei   = (const int*)  d_in[1];
  const int*   batch= (const int*)  d_in[2];
  const float* W1   = (const float*)d_in[3];
  const float* as1  = (const float*)d_in[4];
  const float* ad1  = (const float*)d_in[5];
  const float* b1   = (const float*)d_in[6];
  const float* W2   = (const float*)d_in[7];
  const float* as2  = (const float*)d_in[8];
  const float* ad2  = (const float*)d_in[9];
  const float* b2   = (const float*)d_in[10];
  const float* wfc1 = (const float*)d_in[11];
  const float* bfc1 = (const float*)d_in[12];
  const float* wfc2 = (const float*)d_in[13];
  const float* bfc2 = (const float*)d_in[14];
  const int* src = ei;
  const int* dst = ei + N_EDGES;
  float* out = (float*)d_out;

  // workspace layout (floats)
  float* p     = (float*)d_ws;
  float* h1    = p; p += (size_t)N_NODES * 256;   // [N,4,64]
  float* out1  = p; p += (size_t)N_NODES * 256;
  float* sA1   = p; p += (size_t)N_NODES * 4;
  float* sD1   = p; p += (size_t)N_NODES * 4;
  float* emax  = p; p += (size_t)N_NODES * 4;     // reused by layer 2
  float* denom = p; p += (size_t)N_NODES * 4;     // reused by layer 2
  float* ev    = p; p += (size_t)E_TOT  * 4;      // reused by layer 2
  float* h2    = p; p += (size_t)N_NODES * HID;
  float* out2  = p; p += (size_t)N_NODES * HID;
  float* sA2   = p; p += (size_t)N_NODES;
  float* sD2   = p; p += (size_t)N_NODES;
  float* g     = p; p += (size_t)N_GRAPHS * HID;

  const int TB = 256;
  auto nb = [](long long n, int tb) { return (int)((n + tb - 1) / tb); };

  // ---------------- layer 1 (4 heads, 64 ch) ----------------
  gemm_wmma_f32<IN_DIM, 4 * HID>
      <<<nb((long long)(N_NODES / 16) * (4 * HID / 16) * 32, TB), TB, 0, stream>>>(
          x, W1, h1, N_NODES);
  att_scores<4><<<nb((long long)N_NODES * 4, TB), TB, 0, stream>>>(h1, as1, ad1, sA1, sD1);
  fill_f32<<<nb((long long)N_NODES * 4, TB), TB, 0, stream>>>(emax, -INFINITY, N_NODES * 4);
  fill_f32<<<nb((long long)N_NODES * 4, TB), TB, 0, stream>>>(denom, 0.f, N_NODES * 4);
  fill_f32<<<nb((long long)N_NODES * 256, TB), TB, 0, stream>>>(out1, 0.f, N_NODES * 256);
  edge_lrelu_max<4><<<nb(E_TOT, TB), TB, 0, stream>>>(src, dst, sA1, sD1, emax);
  edge_exp_sum<4><<<nb(E_TOT, TB), TB, 0, stream>>>(src, dst, sA1, sD1, emax, ev, denom);
  edge_msg<4><<<nb((long long)E_TOT * 4 * HID, TB), TB, 0, stream>>>(src, dst, ev, denom, h1, out1);
  bias_relu<256><<<nb((long long)N_NODES * 256, TB), TB, 0, stream>>>(out1, b1, N_NODES);

  // ---------------- layer 2 (1 head, 64 ch) ----------------
  gemm_wmma_f32<4 * HID, HID>
      <<<nb((long long)(N_NODES / 16) * (HID / 16) * 32, TB), TB, 0, stream>>>(
          out1, W2, h2, N_NODES);
  att_scores<1><<<nb(N_NODES, TB), TB, 0, stream>>>(h2, as2, ad2, sA2, sD2);
  fill_f32<<<nb(N_NODES, TB), TB, 0, stream>>>(emax, -INFINITY, N_NODES);
  fill_f32<<<nb(N_NODES, TB), TB, 0, stream>>>(denom, 0.f, N_NODES);
  fill_f32<<<nb((long long)N_NODES * HID, TB), TB, 0, stream>>>(out2, 0.f, N_NODES * HID);
  edge_lrelu_max<1><<<nb(E_TOT, TB), TB, 0, stream>>>(src, dst, sA2, sD2, emax);
  edge_exp_sum<1><<<nb(E_TOT, TB), TB, 0, stream>>>(src, dst, sA2, sD2, emax, ev, denom);
  edge_msg<1><<<nb((long long)E_TOT * HID, TB), TB, 0, stream>>>(src, dst, ev, denom, h2, out2);
  bias_relu<HID><<<nb((long long)N_NODES * HID, TB), TB, 0, stream>>>(out2, b2, N_NODES);

  // ---------------- pool + head ----------------
  fill_f32<<<nb(N_GRAPHS * HID, TB), TB, 0, stream>>>(g, -INFINITY, N_GRAPHS * HID);
  pool_max<<<nb((long long)N_NODES * HID, TB), TB, 0, stream>>>(out2, batch, g);
  head_kernel<<<N_GRAPHS, HID, 0, stream>>>(g, wfc1, bfc1, wfc2, bfc2, out);
}